// RGATBase_53249004536136
// MI455X (gfx1250) — compile-verified
//
#include <hip/hip_runtime.h>
#include <hip/hip_bf16.h>
#include <math.h>

#define NUM_ENT 50000
#define NNODE (4 * NUM_ENT)   // 200000
#define D 128
#define H 4
#define DH 32                  // D/H
#define NE 600000
#define NRELS 20               // 2*NUM_REL
#define NB 4096

typedef float v2f __attribute__((ext_vector_type(2)));
typedef float v8f __attribute__((ext_vector_type(8)));

// ---------- ordered-uint encoding for float atomicMax ----------
__device__ __forceinline__ unsigned f32_key(float f) {
    unsigned b = __float_as_uint(f);
    return (b & 0x80000000u) ? ~b : (b | 0x80000000u);
}
__device__ __forceinline__ float f32_unkey(unsigned k) {
    unsigned b = (k & 0x80000000u) ? (k & 0x7FFFFFFFu) : ~k;
    return __uint_as_float(b);
}

// ---------- build x = concat(id_embed, gender, age, level) ----------
__global__ void build_x_kernel(const int* __restrict__ feat,
                               const float* __restrict__ gw,
                               const float* __restrict__ aw,
                               const float* __restrict__ lw,
                               const float* __restrict__ idemb,
                               float* __restrict__ x) {
    long idx = (long)blockIdx.x * blockDim.x + threadIdx.x;  // one float4
    long total = (long)NNODE * (D / 4);
    if (idx >= total) return;
    int node = (int)(idx / (D / 4));
    int c4 = (int)(idx % (D / 4));
    const float4* srcp;
    if (node < NUM_ENT)
        srcp = (const float4*)(idemb + (size_t)node * D);
    else if (node < 2 * NUM_ENT)
        srcp = (const float4*)(gw + (size_t)feat[(node - NUM_ENT) * 3 + 0] * D);
    else if (node < 3 * NUM_ENT)
        srcp = (const float4*)(aw + (size_t)feat[(node - 2 * NUM_ENT) * 3 + 1] * D);
    else
        srcp = (const float4*)(lw + (size_t)feat[(node - 3 * NUM_ENT) * 3 + 2] * D);
    ((float4*)(x + (size_t)node * D))[c4] = srcp[c4];
}

__global__ void zero_f32(float* __restrict__ p, long n) {
    long i = (long)blockIdx.x * blockDim.x + threadIdx.x;
    if (i < n) p[i] = 0.0f;
}
__global__ void zero_u32(unsigned* __restrict__ p, long n) {
    long i = (long)blockIdx.x * blockDim.x + threadIdx.x;
    if (i < n) p[i] = 0u;   // key(0)=below -inf -> identity for ordered max
}

// ---------- WMMA fp32 GEMM: C[M x 128] = A[M x 128] @ B[128 x 128] ----------
// Block = 256 threads = 8 waves. Block owns 16 rows; wave w owns cols [16w,16w+16).
// Uses V_WMMA_F32_16X16X4_F32, K swept in steps of 4 (full fp32 precision).
__global__ void __launch_bounds__(256)
gemm_wmma_kernel(const float* __restrict__ A, const float* __restrict__ B,
                 float* __restrict__ C) {
    __shared__ float As[16 * 128];
    int tid = threadIdx.x;
    int rowBase = blockIdx.x * 16;

    // cooperative LDS stage of the contiguous 16x128 A tile (8 floats/thread)
    const float4* gsrc = (const float4*)(A + (size_t)rowBase * D);
    float4* ldst = (float4*)As;
    ldst[tid * 2 + 0] = gsrc[tid * 2 + 0];
    ldst[tid * 2 + 1] = gsrc[tid * 2 + 1];
    __syncthreads();

    int wave = tid >> 5;
    int lane = tid & 31;
    int colBase = wave * 16;
    int mrow = lane & 15;          // A: M index held by this lane
    int koff = (lane >> 4) * 2;    // lanes 16-31 hold K+2, K+3
    int ncol = lane & 15;          // B/C: N index held by this lane

    v8f acc = {};
#pragma unroll
    for (int kk = 0; kk < 128; kk += 4) {
        v2f a, b;
        a.x = As[mrow * 128 + kk + koff];
        a.y = As[mrow * 128 + kk + koff + 1];
        b.x = B[(size_t)(kk + koff) * D + colBase + ncol];
        b.y = B[(size_t)(kk + koff + 1) * D + colBase + ncol];
        acc = __builtin_amdgcn_wmma_f32_16x16x4_f32(
            /*neg_a=*/false, a, /*neg_b=*/false, b,
            /*c_mod=*/(short)0, acc, /*reuse_a=*/false, /*reuse_b=*/false);
    }

    // C layout: VGPR r -> row r (lanes 0-15) / row r+8 (lanes 16-31)
    int crow = (lane >> 4) * 8;
    float* cptr = C + (size_t)rowBase * D + colBase + ncol;
#pragma unroll
    for (int r = 0; r < 8; ++r) cptr[(size_t)(crow + r) * D] = acc[r];
}

// ---------- tiny GEMM for relation embeddings (20 x 128 @ 128 x 128) ----------
__global__ void small_gemm_kernel(const float* __restrict__ A,
                                  const float* __restrict__ B,
                                  float* __restrict__ C, int M) {
    int idx = blockIdx.x * blockDim.x + threadIdx.x;
    if (idx >= M * D) return;
    int m = idx >> 7, n = idx & 127;
    float acc = 0.0f;
#pragma unroll 8
    for (int k = 0; k < D; ++k) acc += A[m * D + k] * B[k * D + n];
    C[idx] = acc;
}

// ---------- pass A: per-edge attention scores + segment max ----------
// one wave per edge; lane = head*8 + sub, each lane covers 4 dims
__global__ void __launch_bounds__(256)
edge_score_kernel(const float* __restrict__ xW, const float* __restrict__ rW,
                  const int* __restrict__ src, const int* __restrict__ dst,
                  const int* __restrict__ etype, const float* __restrict__ att,
                  float* __restrict__ score, unsigned* __restrict__ nmax) {
    int wid = (int)(((long)blockIdx.x * blockDim.x + threadIdx.x) >> 5);
    int lane = threadIdx.x & 31;
    if (wid >= NE) return;
    int s = src[wid], d = dst[wid], t = etype[wid];
    int h = lane >> 3, sub = lane & 7;
    int off = h * DH + sub * 4;

    float4 xs = *(const float4*)(xW + (size_t)s * D + off);
    float4 rv = *(const float4*)(rW + (size_t)t * D + off);
    float4 xd = *(const float4*)(xW + (size_t)d * D + off);
    float4 aq = *(const float4*)(att + h * (2 * DH) + sub * 4);
    float4 am = *(const float4*)(att + h * (2 * DH) + DH + sub * 4);

    float mx0 = xs.x + rv.x, mx1 = xs.y + rv.y, mx2 = xs.z + rv.z, mx3 = xs.w + rv.w;
    float p = xd.x * aq.x + xd.y * aq.y + xd.z * aq.z + xd.w * aq.w +
              mx0 * am.x + mx1 * am.y + mx2 * am.z + mx3 * am.w;
    // reduce over the 8 lanes of this head
    p += __shfl_xor(p, 1);
    p += __shfl_xor(p, 2);
    p += __shfl_xor(p, 4);
    float sc = p > 0.0f ? p : 0.2f * p;   // leaky_relu(0.2)
    if (sub == 0) {
        score[(size_t)wid * H + h] = sc;
        atomicMax(&nmax[(size_t)d * H + h], f32_key(sc));
    }
}

// ---------- pass B: exponentiate + segment sum ----------
__global__ void edge_exp_kernel(const int* __restrict__ dst,
                                const unsigned* __restrict__ nmax,
                                float* __restrict__ score,
                                float* __restrict__ nsum) {
    long i = (long)blockIdx.x * blockDim.x + threadIdx.x;
    if (i >= (long)NE * H) return;
    int e = (int)(i >> 2), h = (int)(i & 3);
    int d = dst[e];
    float mx = f32_unkey(nmax[(size_t)d * H + h]);
    if (!(mx > -3.0e38f && mx < 3.0e38f)) mx = 0.0f;  // non-finite -> 0
    float ev = expf(score[i] - mx);
    score[i] = ev;
    atomicAdd(&nsum[(size_t)d * H + h], ev);
}

// ---------- pass C: normalize + scatter-add alpha * m ----------
__global__ void __launch_bounds__(256)
edge_scatter_kernel(const float* __restrict__ xW, const float* __restrict__ rW,
                    const int* __restrict__ src, const int* __restrict__ dst,
                    const int* __restrict__ etype,
                    const float* __restrict__ score,
                    const float* __restrict__ nsum, float* __restrict__ out) {
    int wid = (int)(((long)blockIdx.x * blockDim.x + threadIdx.x) >> 5);
    int lane = threadIdx.x & 31;
    if (wid >= NE) return;
    int s = src[wid], d = dst[wid], t = etype[wid];
    int h = lane >> 3, sub = lane & 7;
    int off = h * DH + sub * 4;

    float ev = score[(size_t)wid * H + h];
    float alpha = ev / (nsum[(size_t)d * H + h] + 1e-16f);

    float4 xs = *(const float4*)(xW + (size_t)s * D + off);
    float4 rv = *(const float4*)(rW + (size_t)t * D + off);
    float* op = out + (size_t)d * D + off;
    atomicAdd(op + 0, alpha * (xs.x + rv.x));
    atomicAdd(op + 1, alpha * (xs.y + rv.y));
    atomicAdd(op + 2, alpha * (xs.z + rv.z));
    atomicAdd(op + 3, alpha * (xs.w + rv.w));
}

__global__ void tanh_kernel(const float* __restrict__ in, float* __restrict__ o,
                            long n4) {
    long i = (long)blockIdx.x * blockDim.x + threadIdx.x;
    if (i >= n4) return;
    float4 v = ((const float4*)in)[i];
    v.x = tanhf(v.x); v.y = tanhf(v.y); v.z = tanhf(v.z); v.w = tanhf(v.w);
    ((float4*)o)[i] = v;
}

__global__ void gather_rows_kernel(const float* __restrict__ table,
                                   const int* __restrict__ idx,
                                   float* __restrict__ o, int nrows) {
    int i = blockIdx.x * blockDim.x + threadIdx.x;
    if (i >= nrows * D) return;
    int r = i >> 7, c = i & 127;
    o[i] = table[(size_t)idx[r] * D + c];
}

extern "C" void kernel_launch(void* const* d_in, const int* in_sizes, int n_in,
                              void* d_out, int out_size, void* d_ws,
                              size_t ws_size, hipStream_t stream) {
    (void)in_sizes; (void)n_in; (void)out_size; (void)ws_size;
    const int* sub          = (const int*)d_in[0];
    const int* rel          = (const int*)d_in[1];
    const int* edge_index   = (const int*)d_in[2];
    const int* edge_type    = (const int*)d_in[3];
    const int* ent_feature  = (const int*)d_in[4];
    const float* gender_w   = (const float*)d_in[5];
    const float* age_w      = (const float*)d_in[6];
    const float* level_w    = (const float*)d_in[7];
    const float* id_embed   = (const float*)d_in[8];
    const float* init_rel   = (const float*)d_in[9];
    const float* W1         = (const float*)d_in[10];
    const float* Wr1        = (const float*)d_in[11];
    const float* att1       = (const float*)d_in[12];
    const float* W2         = (const float*)d_in[13];
    const float* Wr2        = (const float*)d_in[14];
    const float* att2       = (const float*)d_in[15];

    const int* srcI = edge_index;
    const int* dstI = edge_index + NE;

    float* outp    = (float*)d_out;
    float* sub_emb = outp;                      // NB*D
    float* rel_emb = outp + (size_t)NB * D;     // NB*D
    float* x_out   = outp + (size_t)2 * NB * D; // NNODE*D

    // ---- workspace carve ----
    char* ws = (char*)d_ws;
    size_t o = 0;
    auto carve = [&](size_t bytes) -> void* {
        void* p = ws + o;
        o += (bytes + 255) & ~(size_t)255;
        return p;
    };
    float*    xbuf   = (float*)carve((size_t)NNODE * D * sizeof(float));
    float*    xWbuf  = (float*)carve((size_t)NNODE * D * sizeof(float));
    float*    outbuf = (float*)carve((size_t)NNODE * D * sizeof(float));
    float*    score  = (float*)carve((size_t)NE * H * sizeof(float));
    unsigned* nmax   = (unsigned*)carve((size_t)NNODE * H * sizeof(unsigned));
    float*    nsum   = (float*)carve((size_t)NNODE * H * sizeof(float));
    float*    rW     = (float*)carve((size_t)NRELS * D * sizeof(float));
    float*    r1     = (float*)carve((size_t)NRELS * D * sizeof(float));
    float*    r2     = (float*)carve((size_t)NRELS * D * sizeof(float));

    const int TB = 256;
    long nx4 = (long)NNODE * (D / 4);

    // x = concat(...)
    build_x_kernel<<<(unsigned)((nx4 + TB - 1) / TB), TB, 0, stream>>>(
        ent_feature, gender_w, age_w, level_w, id_embed, xbuf);

    // relation transforms for layer 1
    small_gemm_kernel<<<(NRELS * D + TB - 1) / TB, TB, 0, stream>>>(init_rel, W1, rW, NRELS);
    small_gemm_kernel<<<(NRELS * D + TB - 1) / TB, TB, 0, stream>>>(init_rel, Wr1, r1, NRELS);

    for (int layer = 0; layer < 2; ++layer) {
        const float* Wl   = (layer == 0) ? W1 : W2;
        const float* attl = (layer == 0) ? att1 : att2;
        float* xdst = (layer == 0) ? xbuf : x_out;  // tanh destination

        long nh = (long)NNODE * H;
        zero_u32<<<(unsigned)((nh + TB - 1) / TB), TB, 0, stream>>>(nmax, nh);
        zero_f32<<<(unsigned)((nh + TB - 1) / TB), TB, 0, stream>>>(nsum, nh);
        long nd = (long)NNODE * D;
        zero_f32<<<(unsigned)((nd + TB - 1) / TB), TB, 0, stream>>>(outbuf, nd);

        // xW = x @ W  (WMMA fp32)
        gemm_wmma_kernel<<<NNODE / 16, TB, 0, stream>>>(xbuf, Wl, xWbuf);

        // attention: scores -> softmax -> scatter
        unsigned eb = (NE * 32 + TB - 1) / TB;  // wave per edge
        edge_score_kernel<<<eb, TB, 0, stream>>>(xWbuf, rW, srcI, dstI, edge_type,
                                                 attl, score, nmax);
        long neh = (long)NE * H;
        edge_exp_kernel<<<(unsigned)((neh + TB - 1) / TB), TB, 0, stream>>>(
            dstI, nmax, score, nsum);
        edge_scatter_kernel<<<eb, TB, 0, stream>>>(xWbuf, rW, srcI, dstI, edge_type,
                                                   score, nsum, outbuf);
        // x = tanh(out)
        tanh_kernel<<<(unsigned)((nx4 + TB - 1) / TB), TB, 0, stream>>>(outbuf, xdst, nx4);

        if (layer == 0) {
            // relation transforms for layer 2 (r1 = init_rel @ Wr1)
            small_gemm_kernel<<<(NRELS * D + TB - 1) / TB, TB, 0, stream>>>(r1, W2, rW, NRELS);
            small_gemm_kernel<<<(NRELS * D + TB - 1) / TB, TB, 0, stream>>>(r1, Wr2, r2, NRELS);
        }
    }

    // final gathers
    gather_rows_kernel<<<(NB * D + TB - 1) / TB, TB, 0, stream>>>(x_out, sub, sub_emb, NB);
    gather_rows_kernel<<<(NB * D + TB - 1) / TB, TB, 0, stream>>>(r2, rel, rel_emb, NB);
}